// PosAttn_TokenOut_21534966022583
// MI455X (gfx1250) — compile-verified
//
#include <hip/hip_runtime.h>
#include <hip/hip_bf16.h>
#include <math.h>

// Problem constants (from reference)
#define VV    32000
#define LL    2048
#define BB    2
#define NROWS (BB * (LL - 1))   // 4094 output rows
#define NROWS_PAD 4096          // padded to multiple of 16

typedef __attribute__((ext_vector_type(2))) float v2f;
typedef __attribute__((ext_vector_type(8))) float v8f;

__device__ __forceinline__ float softplus_f(float z) {
    return fmaxf(z, 0.0f) + log1pf(__expf(-fabsf(z)));
}

// ---------------------------------------------------------------------------
// Phase 1: per-row online softmax attention + E-gather accumulate, fold tau/WE.
// One 256-thread block per row i (i = blockIdx.x + 1; row 0 dropped by slice).
// Writes dvec[row][0..3], row = b*2047 + (i-1); dvec[row][3] = 0 (K pad).
// ---------------------------------------------------------------------------
__global__ __launch_bounds__(256) void attn_dvec_kernel(
    const int*   __restrict__ x,     // [B, L]
    const float* __restrict__ E,     // [V, 3]
    const float* __restrict__ Ppos,  // [L, 2]
    const float* __restrict__ Mp,    // [2, 2]
    const float* __restrict__ rbeta, // scalar
    const float* __restrict__ rtau,  // scalar
    const float* __restrict__ WE,    // [3, 3]
    float*       __restrict__ dvec)  // [NROWS_PAD, 4]
{
    const int i   = blockIdx.x + 1;      // row 1..2047
    const int tid = threadIdx.x;

    const float beta = softplus_f(rbeta[0]) + 1e-6f;
    const float tau  = softplus_f(rtau[0])  + 1e-6f;

    // Fold Mp into the query: score(j) = qx*pj.x + qy*pj.y
    const float pix = Ppos[2 * i + 0], piy = Ppos[2 * i + 1];
    const float qx = beta * (pix * Mp[0] + piy * Mp[2]);
    const float qy = beta * (pix * Mp[1] + piy * Mp[3]);

    // Online softmax with fused gather-accumulate of E[x[b,j]]
    float m = -INFINITY, s = 0.0f;
    float acc[6] = {0.f, 0.f, 0.f, 0.f, 0.f, 0.f};

    for (int j = tid; j < i; j += 256) {
        const float pjx = Ppos[2 * j + 0], pjy = Ppos[2 * j + 1];
        const float sc  = qx * pjx + qy * pjy;
        const float nm    = fmaxf(m, sc);
        const float scale = __expf(m - nm);    // first iter: exp(-inf) = 0
        const float w     = __expf(sc - nm);
        s = s * scale + w;
        const int t0 = x[j];
        const int t1 = x[LL + j];
        const float* e0 = E + (size_t)t0 * 3;
        const float* e1 = E + (size_t)t1 * 3;
        acc[0] = acc[0] * scale + w * e0[0];
        acc[1] = acc[1] * scale + w * e0[1];
        acc[2] = acc[2] * scale + w * e0[2];
        acc[3] = acc[3] * scale + w * e1[0];
        acc[4] = acc[4] * scale + w * e1[1];
        acc[5] = acc[5] * scale + w * e1[2];
        m = nm;
    }

    __shared__ float sm[256], ss[256], sa[6][256];
    sm[tid] = m; ss[tid] = s;
#pragma unroll
    for (int e = 0; e < 6; ++e) sa[e][tid] = acc[e];
    __syncthreads();

    for (int off = 128; off > 0; off >>= 1) {
        if (tid < off) {
            const float m1 = sm[tid], m2 = sm[tid + off];
            const float s1 = ss[tid], s2 = ss[tid + off];
            const float M  = fmaxf(m1, m2);
            const float w1 = (s1 > 0.0f) ? __expf(m1 - M) : 0.0f;
            const float w2 = (s2 > 0.0f) ? __expf(m2 - M) : 0.0f;
            sm[tid] = M;
            ss[tid] = s1 * w1 + s2 * w2;
#pragma unroll
            for (int e = 0; e < 6; ++e)
                sa[e][tid] = sa[e][tid] * w1 + sa[e][tid + off] * w2;
        }
        __syncthreads();
    }

    if (tid == 0) {
        const float invS = 1.0f / ss[0];
#pragma unroll
        for (int b = 0; b < BB; ++b) {
            const float c0 = sa[b * 3 + 0][0] * invS;
            const float c1 = sa[b * 3 + 1][0] * invS;
            const float c2 = sa[b * 3 + 2][0] * invS;
            float* dp = dvec + (size_t)(b * (LL - 1) + (i - 1)) * 4;
            dp[0] = tau * (c0 * WE[0] + c1 * WE[3] + c2 * WE[6]);
            dp[1] = tau * (c0 * WE[1] + c1 * WE[4] + c2 * WE[7]);
            dp[2] = tau * (c0 * WE[2] + c1 * WE[5] + c2 * WE[8]);
            dp[3] = 0.0f;                          // K=4 pad lane
        }
    }
    // zero the two pad rows (only block 0)
    if (blockIdx.x == 0 && tid < 8)
        dvec[(size_t)NROWS * 4 + tid] = 0.0f;
}

// ---------------------------------------------------------------------------
// Phase 2: logits[row, v] = dvec[row,:] . Epad[v,:]  via V_WMMA_F32_16X16X4_F32
// One wave per 16-row x 64-col strip (4 WMMA tiles, A fragment reused).
//
// VGPR layouts (CDNA5 ISA §7.12.2, 32-bit, wave32):
//   A 16x4 : M = lane&15 (both halves); VGPR0 = K(2*hi), VGPR1 = K(2*hi+1)
//   B 4x16 : N = lane&15; VGPR0 = B[2*hi][n], VGPR1 = B[2*hi+1][n]
//   D 16x16: VGPR r -> row (r + 8*hi), col = lane&15
//
// Stores: single per-lane base address, 8 NT stores with immediate offsets
// r * V * 4 bytes (max 896000 < 2^23, fits the signed 24-bit IOFFSET field).
// Only the last row tile (contains pad rows 4094/4095) takes the predicated
// edge path.
// ---------------------------------------------------------------------------
__global__ __launch_bounds__(256) void logits_wmma_kernel(
    const float* __restrict__ E,     // [V, 3]
    const float* __restrict__ dvec,  // [NROWS_PAD, 4]
    float*       __restrict__ out)   // [NROWS, V]
{
    const int lane = threadIdx.x & 31;
    const int wave = threadIdx.x >> 5;
    const int n    = lane & 15;
    const int hi   = lane >> 4;

    const int rowTile = blockIdx.y << 4;              // 0..4080
    const int vgroup  = blockIdx.x * 8 + wave;        // 64 columns per group
    const int vbase0  = vgroup * 64;
    if (vbase0 >= VV) return;                         // wave-uniform exit

    // A fragment: row = rowTile + (lane&15); K pair selected by half-wave
    const float* ap = dvec + (size_t)(rowTile + n) * 4 + 2 * hi;
    v2f a;
    a.x = ap[0];
    a.y = ap[1];

    const int rbase = rowTile + hi * 8;
    // Per-lane output base for tile t=0 (column vbase0 + n, row rbase)
    float* obase = out + (size_t)rbase * VV + vbase0 + n;

    if (rowTile + 16 <= NROWS) {
        // ------- fast path: full tile, straight-line stores -------
#pragma unroll
        for (int t = 0; t < 4; ++t) {
            const int vbase = vbase0 + t * 16;
            const float* ep = E + (size_t)(vbase + n) * 3;
            // load all 3 row floats unconditionally, select via cndmask
            const float e0v = ep[0], e1v = ep[1], e2v = ep[2];
            v2f b;
            b.x = hi ? e2v : e0v;                     // K = 2*hi
            b.y = hi ? 0.0f : e1v;                    // K = 2*hi+1 (K=3 -> 0)
            if (t < 3)
                __builtin_prefetch(E + (size_t)(vbase + 16 + n) * 3, 0, 3);

            v8f c = {0.f, 0.f, 0.f, 0.f, 0.f, 0.f, 0.f, 0.f};
            v8f dmat = __builtin_amdgcn_wmma_f32_16x16x4_f32(
                false, a, false, b, (short)0, c, false, false);

            float* op = obase + t * 16;
#pragma unroll
            for (int r = 0; r < 8; ++r)
                __builtin_nontemporal_store(dmat[r], op + (size_t)r * VV);
        }
    } else {
        // ------- edge path: last row tile only (rows 4094/4095 are pad) ----
#pragma unroll
        for (int t = 0; t < 4; ++t) {
            const int vbase = vbase0 + t * 16;
            const float* ep = E + (size_t)(vbase + n) * 3;
            const float e0v = ep[0], e1v = ep[1], e2v = ep[2];
            v2f b;
            b.x = hi ? e2v : e0v;
            b.y = hi ? 0.0f : e1v;

            v8f c = {0.f, 0.f, 0.f, 0.f, 0.f, 0.f, 0.f, 0.f};
            v8f dmat = __builtin_amdgcn_wmma_f32_16x16x4_f32(
                false, a, false, b, (short)0, c, false, false);

            float* op = obase + t * 16;
#pragma unroll
            for (int r = 0; r < 8; ++r)
                if (rbase + r < NROWS)
                    __builtin_nontemporal_store(dmat[r], op + (size_t)r * VV);
        }
    }
}

extern "C" void kernel_launch(void* const* d_in, const int* in_sizes, int n_in,
                              void* d_out, int out_size, void* d_ws, size_t ws_size,
                              hipStream_t stream) {
    const int*   x     = (const int*)  d_in[0];
    const float* E     = (const float*)d_in[1];
    const float* Ppos  = (const float*)d_in[2];
    const float* Mp    = (const float*)d_in[3];
    const float* rbeta = (const float*)d_in[4];
    const float* rtau  = (const float*)d_in[5];
    const float* WE    = (const float*)d_in[6];
    float* out  = (float*)d_out;
    float* dvec = (float*)d_ws;          // NROWS_PAD * 4 floats = 64 KB

    attn_dvec_kernel<<<LL - 1, 256, 0, stream>>>(x, E, Ppos, Mp, rbeta, rtau, WE, dvec);

    // 500 v-groups of 64 cols; 8 waves/block -> 63 blocks in x; 256 row tiles
    dim3 grid((500 + 7) / 8, NROWS_PAD / 16);
    logits_wmma_kernel<<<grid, 256, 0, stream>>>(E, dvec, out);
}